// SimpleMPNNLayer_40604620817079
// MI455X (gfx1250) — compile-verified
//
#include <hip/hip_runtime.h>

#define N_NODES 100000
#define D_FEAT  128
#define N_EDGES 640000
#define NT_TILES (N_NODES / 16)   // 6250, exact

typedef float v2f __attribute__((ext_vector_type(2)));
typedef float v8f __attribute__((ext_vector_type(8)));

// ---------------------------------------------------------------------------
// Kernel 1: zero agg[N*D] and deg[N] in workspace
// ---------------------------------------------------------------------------
__global__ void zero_ws_kernel(float4* __restrict__ ws, int n4) {
    int i = blockIdx.x * blockDim.x + threadIdx.x;
    if (i < n4) ws[i] = make_float4(0.f, 0.f, 0.f, 0.f);
}

// ---------------------------------------------------------------------------
// Kernel 2: edge scatter. One wave32 per edge; lane l handles floats [4l,4l+4).
// x row read as coalesced float4; hardware f32 atomic adds into agg (L2).
// ---------------------------------------------------------------------------
__global__ __launch_bounds__(256) void edge_scatter_kernel(
    const long long* __restrict__ ei, const float* __restrict__ x,
    float* __restrict__ agg, float* __restrict__ deg) {
    int e = (blockIdx.x * blockDim.x + threadIdx.x) >> 5;
    if (e >= N_EDGES) return;
    int lane = threadIdx.x & 31;
    long long src = ei[e];
    long long dst = ei[N_EDGES + e];
    const float4 v = *(const float4*)(x + src * D_FEAT + lane * 4);
    float* a = agg + dst * D_FEAT + lane * 4;
    unsafeAtomicAdd(a + 0, v.x);
    unsafeAtomicAdd(a + 1, v.y);
    unsafeAtomicAdd(a + 2, v.z);
    unsafeAtomicAdd(a + 3, v.w);
    if (lane == 0) unsafeAtomicAdd(deg + dst, 1.0f);
}

// ---------------------------------------------------------------------------
// Kernel 3: fused  out = LN(gelu(x@Ws^T + bs + (agg/deg)@Wn^T + bn))
// One wave per 16-row tile, f32 WMMA 16x16x4.
//   A frag (16x4 f32): lane m=lane&15 holds row m; half=lane>>4 selects K pair
//   B frag (4x16 f32): B[k][j] = w[j][k] -> lane holds column j=lane&15
//   C/D (16x16 f32, 8 VGPRs): row = v + 8*half, col = lane&15
// ---------------------------------------------------------------------------
__global__ __launch_bounds__(128) void mpnn_fused_kernel(
    const float* __restrict__ x, const float* __restrict__ agg,
    const float* __restrict__ deg,
    const float* __restrict__ wself, const float* __restrict__ bself,
    const float* __restrict__ wnei,  const float* __restrict__ bnei,
    const float* __restrict__ gamma, const float* __restrict__ beta,
    float* __restrict__ out) {
    int tile = blockIdx.x * 4 + (threadIdx.x >> 5);
    if (tile >= NT_TILES) return;                 // wave-uniform: EXEC stays all-1
    int lane = threadIdx.x & 31;
    int half = lane >> 4;                         // 0 or 1
    int m    = lane & 15;
    int rowA = tile * 16 + m;

    float dv   = deg[rowA];
    float rdeg = 1.0f / fmaxf(dv, 1.0f);

    // Preload A fragments for x and normalized agg (read each row exactly once)
    v2f xa[32], ga[32];
    const float* xr = x   + (long)rowA * D_FEAT + 2 * half;
    const float* gr = agg + (long)rowA * D_FEAT + 2 * half;
#pragma unroll
    for (int s = 0; s < 32; ++s) {
        xa[s] = *(const v2f*)(xr + 4 * s);
        v2f g = *(const v2f*)(gr + 4 * s);
        ga[s] = g * rdeg;
    }

    // GEMM: accumulate both weight matrices into the same C tile
    v8f acc[8];
#pragma unroll
    for (int nt = 0; nt < 8; ++nt) {
        int col = nt * 16 + m;
        const float* ws = wself + (long)col * D_FEAT + 2 * half;
        const float* wn = wnei  + (long)col * D_FEAT + 2 * half;
        v8f c = {};
#pragma unroll
        for (int s = 0; s < 32; ++s) {
            v2f bs = *(const v2f*)(ws + 4 * s);
            c = __builtin_amdgcn_wmma_f32_16x16x4_f32(
                    false, xa[s], false, bs, (short)0, c, false, false);
            v2f bn = *(const v2f*)(wn + 4 * s);
            c = __builtin_amdgcn_wmma_f32_16x16x4_f32(
                    false, ga[s], false, bn, (short)0, c, false, false);
        }
        acc[nt] = c;
    }

    // Bias + exact-erf GELU, accumulate row sums for LayerNorm
    float s1[8], s2[8];
#pragma unroll
    for (int v = 0; v < 8; ++v) { s1[v] = 0.f; s2[v] = 0.f; }
#pragma unroll
    for (int nt = 0; nt < 8; ++nt) {
        int col = nt * 16 + m;
        float bias = bself[col] + bnei[col];
#pragma unroll
        for (int v = 0; v < 8; ++v) {
            float h = acc[nt][v] + bias;
            h = 0.5f * h * (1.0f + erff(h * 0.70710678118654752f));
            acc[nt][v] = h;
            s1[v] += h;
            s2[v] += h * h;
        }
    }

    // Cross-lane reduce within each 16-lane half (rows stay within a half)
#pragma unroll
    for (int v = 0; v < 8; ++v) {
#pragma unroll
        for (int msk = 1; msk < 16; msk <<= 1) {
            s1[v] += __shfl_xor(s1[v], msk, 32);
            s2[v] += __shfl_xor(s2[v], msk, 32);
        }
    }
    float mu[8], rs[8];
#pragma unroll
    for (int v = 0; v < 8; ++v) {
        float mean = s1[v] * (1.0f / 128.0f);
        float var  = s2[v] * (1.0f / 128.0f) - mean * mean;
        mu[v] = mean;
        rs[v] = rsqrtf(var + 1e-5f);
    }

    // Scale/shift and store
#pragma unroll
    for (int nt = 0; nt < 8; ++nt) {
        int col = nt * 16 + m;
        float gm = gamma[col], bt = beta[col];
#pragma unroll
        for (int v = 0; v < 8; ++v) {
            int row = tile * 16 + v + 8 * half;
            out[(long)row * D_FEAT + col] = (acc[nt][v] - mu[v]) * rs[v] * gm + bt;
        }
    }
}

// ---------------------------------------------------------------------------
extern "C" void kernel_launch(void* const* d_in, const int* in_sizes, int n_in,
                              void* d_out, int out_size, void* d_ws, size_t ws_size,
                              hipStream_t stream) {
    const float*     x     = (const float*)d_in[0];
    const long long* ei    = (const long long*)d_in[1];
    const float*     wself = (const float*)d_in[2];
    const float*     bself = (const float*)d_in[3];
    const float*     wnei  = (const float*)d_in[4];
    const float*     bnei  = (const float*)d_in[5];
    const float*     gamma = (const float*)d_in[6];
    const float*     beta  = (const float*)d_in[7];
    float*           out   = (float*)d_out;

    float* agg = (float*)d_ws;                               // N*D floats
    float* deg = agg + (size_t)N_NODES * D_FEAT;             // N floats

    int n4 = (N_NODES * D_FEAT + N_NODES) / 4;               // both divisible by 4
    zero_ws_kernel<<<(n4 + 255) / 256, 256, 0, stream>>>((float4*)d_ws, n4);

    edge_scatter_kernel<<<N_EDGES / 8, 256, 0, stream>>>(ei, x, agg, deg);

    mpnn_fused_kernel<<<(NT_TILES + 3) / 4, 128, 0, stream>>>(
        x, agg, deg, wself, bself, wnei, bnei, gamma, beta, out);
}